// PseudoGroupContrast_65506841198977
// MI455X (gfx1250) — compile-verified
//
#include <hip/hip_runtime.h>
#include <math.h>

// Problem constants (from reference)
#define PROJ   128
#define BATCH  128
#define CLS    50
#define QS     4096
#define NQ     (CLS * QS)          // 204800 queue rows
#define QROWS_PER_BLOCK 128        // queue rows per block (16 per wave x 8 waves)
#define NBLK2  (NQ / QROWS_PER_BLOCK)  // 1600 blocks
#define LDS_STRIDE 132             // padded f row stride (floats): conflict-free b64
#define LDS_BYTES  (BATCH * LDS_STRIDE * 4)   // 67584 B (CDNA5 WGP has 320KB)
#define EPS_F  1e-8f

typedef float v2f __attribute__((ext_vector_type(2)));
typedef float v8f __attribute__((ext_vector_type(8)));

#if defined(__has_builtin)
#  if __has_builtin(__builtin_amdgcn_global_load_async_to_lds_b128)
#    define HAVE_ASYNC_LDS 1
#  endif
#  if __has_builtin(__builtin_amdgcn_s_wait_asynccnt)
#    define HAVE_WAIT_ASYNC_BUILTIN 1
#  endif
#endif

#if defined(HAVE_ASYNC_LDS)
// Parameter types per hipcc diagnostic: global/LDS pointers to a 16-byte
// int vector ("int __attribute__((vector_size(16))) __device__ *").
typedef int b128_t __attribute__((vector_size(16)));
typedef __attribute__((address_space(1))) b128_t* gas_b128;
typedef __attribute__((address_space(3))) b128_t* las_b128;
#endif

// ---------------------------------------------------------------------------
// Block-wide sum for blocks of 128 threads (4 waves of 32, wave32).
// ---------------------------------------------------------------------------
__device__ __forceinline__ float blockSum128(float v, float* sbuf) {
  #pragma unroll
  for (int m = 16; m >= 1; m >>= 1) v += __shfl_xor(v, m, 32);
  const int lane = threadIdx.x & 31;
  const int wv   = threadIdx.x >> 5;
  if (lane == 0) sbuf[wv] = v;
  __syncthreads();
  float r = sbuf[0] + sbuf[1] + sbuf[2] + sbuf[3];
  __syncthreads();
  return r;
}

// ---------------------------------------------------------------------------
// Kernel 1: L2-normalize activation rows, compute l_pos[i] = <f_i, ef_i>.
// ---------------------------------------------------------------------------
__global__ __launch_bounds__(PROJ)
void prep_kernel(const float* __restrict__ act, const float* __restrict__ ema,
                 float* __restrict__ fnorm, float* __restrict__ lpos) {
  __shared__ float sbuf[4];
  const int i = blockIdx.x;
  const int t = threadIdx.x;
  const float a = act[i * PROJ + t];
  const float e = ema[i * PROJ + t];
  const float sa = blockSum128(a * a, sbuf);
  const float se = blockSum128(e * e, sbuf);
  const float fa = a / fmaxf(sqrtf(sa), 1e-12f);
  const float fe = e / fmaxf(sqrtf(se), 1e-12f);
  fnorm[i * PROJ + t] = fa;
  const float lp = blockSum128(fa * fe, sbuf);
  if (t == 0) lpos[i] = lp;
}

// ---------------------------------------------------------------------------
// Kernel 2: sims = queue @ f^T via V_WMMA_F32_16X16X4_F32, fused with
// exp(sims/T) reduction over queue rows.
//
// Role swap vs the naive layout: A = 16 queue rows (unique per wave!), so
// every queue element is loaded from global exactly once grid-wide, and one
// global b64 A-fragment feeds 8 WMMAs (one per batch N-tile). f (64KB) is
// staged once per block into LDS (async global->LDS if available) with row
// stride 132 floats -> ds_load_b64 B fragments are conflict-free.
//
// Fragment layouts (ISA 7.12.2, 32-bit A 16x4 / B 4x16):
//   lanes  0-15: (row=lane,    K = 4k + {0,1}) in VGPR{0,1}
//   lanes 16-31: (row=lane-16, K = 4k + {2,3}) in VGPR{0,1}
// C 16x16: lane%16 = N (batch col), VGPR v + 8*(lane/16) = M (queue row).
// ---------------------------------------------------------------------------
__global__ __launch_bounds__(256)
void gemm_expsum_kernel(const float* __restrict__ fnorm,
                        const float* __restrict__ queue,
                        float* __restrict__ pp) {
  extern __shared__ __align__(16) float ldsf[];   // BATCH * LDS_STRIDE floats
  __shared__ float wsum[8][BATCH];                // per-wave batch-row partials

  const int tid  = threadIdx.x;
  const int lane = tid & 31;
  const int wave = tid >> 5;                 // 0..7
  const int l16  = lane & 15;
  const int hi   = lane >> 4;                // lane half
  const int koff = hi ? 2 : 0;

  // ---- Stage f (128x128) into LDS with padded stride ----
  for (int idx = tid; idx < BATCH * (PROJ / 4); idx += 256) {
    const int r  = idx >> 5;                 // batch row 0..127
    const int c4 = idx & 31;                 // float4 chunk 0..31
#if defined(HAVE_ASYNC_LDS)
    __builtin_amdgcn_global_load_async_to_lds_b128(
        (gas_b128)(fnorm + r * PROJ + 4 * c4),
        (las_b128)(ldsf + r * LDS_STRIDE + 4 * c4),
        0, 0);
#else
    const float4 q = *(const float4*)(fnorm + r * PROJ + 4 * c4);
    *(float4*)&ldsf[r * LDS_STRIDE + 4 * c4] = q;  // (132r+4c)*4 is 16B aligned
#endif
  }
#if defined(HAVE_ASYNC_LDS)
#  if defined(HAVE_WAIT_ASYNC_BUILTIN)
  __builtin_amdgcn_s_wait_asynccnt(0);
#  else
  asm volatile("s_wait_asynccnt 0" ::: "memory");
#  endif
#endif
  __syncthreads();

  // ---- Main GEMM: wave owns queue rows [qbase, qbase+16) ----
  const size_t qrow = (size_t)blockIdx.x * QROWS_PER_BLOCK + wave * 16 + l16;
  const float* aptr = queue + qrow * PROJ + koff;

  v8f c[8] = {};
  #pragma unroll
  for (int k = 0; k < 32; ++k) {
    if ((k & 7) == 0)
      __builtin_prefetch(aptr + 4 * k + 64, 0, 1);   // global_prefetch_b8
    const v2f a = *(const v2f*)(aptr + 4 * k);       // unique queue fragment
    #pragma unroll
    for (int nt = 0; nt < 8; ++nt) {
      const v2f b = *(const v2f*)(ldsf + (nt * 16 + l16) * LDS_STRIDE + 4 * k + koff);
      c[nt] = __builtin_amdgcn_wmma_f32_16x16x4_f32(
                  /*neg_a=*/false, a,
                  /*neg_b=*/false, b,
                  /*c_mod=*/(short)0, c[nt],
                  /*reuse_a=*/false, /*reuse_b=*/false);
    }
  }

  // ---- exp(2*s) reduced over this wave's 16 queue rows, per batch col ----
  #pragma unroll
  for (int nt = 0; nt < 8; ++nt) {
    float bs = 0.f;
    #pragma unroll
    for (int v = 0; v < 8; ++v)
      bs += __expf(2.0f * c[nt][v]);     // rows v (half0) / v+8 (half1)
    bs += __shfl_xor(bs, 16, 32);        // combine lane halves -> all 16 rows
    if (hi == 0) wsum[wave][nt * 16 + l16] = bs;
  }
  __syncthreads();

  // ---- combine the 8 waves -> one partial per batch row (deterministic) ----
  if (tid < BATCH) {
    float s = 0.f;
    #pragma unroll
    for (int w = 0; w < 8; ++w) s += wsum[w][tid];
    pp[(size_t)blockIdx.x * BATCH + tid] = s;
  }
}

// ---------------------------------------------------------------------------
// Kernel 3: per batch row i: finish denom, then scan the 4096 queue entries
// of class pseudo_label[i] (L2-resident: whole queue (105MB) < 192MB L2) for
//   pos = w_i * sum_j qw_j * -log(exp(2 s_ij)/denom + EPS)
// ---------------------------------------------------------------------------
__global__ __launch_bounds__(256)
void pos_kernel(const float* __restrict__ fnorm, const float* __restrict__ lpos,
                const float* __restrict__ pp, const float* __restrict__ w,
                const float* __restrict__ qw, const int* __restrict__ label,
                const float* __restrict__ queue,
                float* __restrict__ per_sample) {
  __shared__ float frow[PROJ];
  __shared__ float red[256];
  const int i = blockIdx.x;
  const int t = threadIdx.x;

  if (t < PROJ) frow[t] = fnorm[i * PROJ + t];

  // denom = l_pos[i] + sum over NBLK2 block partials
  float dp = 0.f;
  for (int b = t; b < NBLK2; b += 256) dp += pp[(size_t)b * BATCH + i];
  red[t] = dp;
  __syncthreads();
  for (int s = 128; s > 0; s >>= 1) {
    if (t < s) red[t] += red[t + s];
    __syncthreads();
  }
  const float lpi   = lpos[i];
  const float denom = lpi + red[0];
  __syncthreads();
  const float invd = 1.0f / denom;

  const int cls = label[i];
  const size_t base = (size_t)cls * QS;
  float partial = 0.f;
  for (int jj = t; jj < QS; jj += 256) {
    const size_t j = base + jj;
    const float4* qr = (const float4*)(queue + j * PROJ);
    float s = 0.f;
    #pragma unroll
    for (int k = 0; k < PROJ / 4; ++k) {
      const float4 q = qr[k];
      s += q.x * frow[4 * k + 0] + q.y * frow[4 * k + 1] +
           q.z * frow[4 * k + 2] + q.w * frow[4 * k + 3];
    }
    const float e = __expf(2.0f * s);
    partial += qw[j] * (-__logf(e * invd + EPS_F));
  }
  red[t] = partial;
  __syncthreads();
  for (int s = 128; s > 0; s >>= 1) {
    if (t < s) red[t] += red[t + s];
    __syncthreads();
  }
  if (t == 0) {
    const float wi   = w[i];
    const float pos  = wi * red[0];
    const float emaT = wi * (-__logf(lpi * invd + EPS_F));
    per_sample[i] = (emaT + pos) / (float)(QS + 1);
  }
}

// ---------------------------------------------------------------------------
// Kernel 4: mean over the 128 per-sample values.
// ---------------------------------------------------------------------------
__global__ __launch_bounds__(BATCH)
void mean_kernel(const float* __restrict__ per_sample, float* __restrict__ out) {
  __shared__ float sbuf[4];
  const float v = per_sample[threadIdx.x];
  const float s = blockSum128(v, sbuf);
  if (threadIdx.x == 0) out[0] = s / (float)BATCH;
}

// ---------------------------------------------------------------------------
// Launch
// ---------------------------------------------------------------------------
extern "C" void kernel_launch(void* const* d_in, const int* in_sizes, int n_in,
                              void* d_out, int out_size, void* d_ws, size_t ws_size,
                              hipStream_t stream) {
  const float* act   = (const float*)d_in[0];  // (128,128)
  const float* ema   = (const float*)d_in[1];  // (128,128)
  const float* w     = (const float*)d_in[2];  // (128,1)
  const float* queue = (const float*)d_in[3];  // (204800,128)
  const float* qw    = (const float*)d_in[4];  // (204800,1)
  const int*   label = (const int*)d_in[5];    // (128,)

  float* ws = (float*)d_ws;
  float* fnorm      = ws;                        // 128*128
  float* lpos       = fnorm + BATCH * PROJ;      // 128
  float* per_sample = lpos + BATCH;              // 128
  float* pp         = per_sample + BATCH;        // NBLK2*128 block partials

  prep_kernel<<<BATCH, PROJ, 0, stream>>>(act, ema, fnorm, lpos);
  gemm_expsum_kernel<<<NBLK2, 256, LDS_BYTES, stream>>>(fnorm, queue, pp);
  pos_kernel<<<BATCH, 256, 0, stream>>>(fnorm, lpos, pp, w, qw, label, queue,
                                        per_sample);
  mean_kernel<<<1, BATCH, 0, stream>>>(per_sample, (float*)d_out);
}